// PCENCompression_46505905881165
// MI455X (gfx1250) — compile-verified
//
#include <hip/hip_runtime.h>
#include <stdint.h>

#define B_     64
#define C_     128
#define T_     8000
#define TILE_T 64                 // timesteps staged per tile
#define NT     (T_ / TILE_T)      // 125 tiles (exact)
#define CHUNKS (TILE_T / 16)      // 4 WMMA chunks per tile
#define WPB    4                  // waves per block
#define NBUF   4                  // async pipeline depth (tiles in flight)
#define EPSF   1e-6f

typedef float v2f __attribute__((ext_vector_type(2)));
typedef float v8f __attribute__((ext_vector_type(8)));

// One wave owns 16 rows (same channel c, batches b0..b0+15) and walks T in
// 64-t tiles. Scan-as-matmul: SM(16x16) = L(16x16) * X(16x16) + a^{i+1}*carry,
// L[i][k] = (k<=i) ? s*a^{i-k} : 0, done as 4 chained V_WMMA_F32_16X16X4_F32.
// x streams to LDS via global_load_async_to_lds (4-tile pipeline, ASYNCcnt);
// PCEN results overwrite x in LDS, then store out coalesced (b128, 2 rows/instr).
__global__ __launch_bounds__(WPB * 32) void pcen_wmma_kernel(
    const float* __restrict__ x,
    const float* __restrict__ alpha_p,
    const float* __restrict__ delta_p,
    const float* __restrict__ r_p,
    const float* __restrict__ s_p,
    float* __restrict__ out)
{
    __shared__ float lds[WPB][NBUF][16 * TILE_T];   // 64 KB / block

    const int lane = threadIdx.x & 31;
    const int wv   = threadIdx.x >> 5;
    const int task = blockIdx.x * WPB + wv;         // 0..511
    const int c    = task >> 2;                     // channel (shared scan matrix)
    const int b0   = (task & 3) * 16;               // first batch row of this wave

    // ---- per-channel params (clipped as in reference) ----
    const float s  = fminf(fmaxf(s_p[c],     0.001f), 0.5f);
    const float a  = 1.0f - s;
    const float al = fminf(fmaxf(alpha_p[c], 0.5f),   1.0f);
    const float de = fminf(fmaxf(delta_p[c], 0.5f),   10.0f);
    const float rr = fminf(fmaxf(r_p[c],     0.1f),   1.0f);
    const float lg2a = __builtin_amdgcn_logf(a);                          // log2(a)
    const float der  = __builtin_amdgcn_exp2f(rr * __builtin_amdgcn_logf(de)); // delta^r

    const int hi = lane >> 4;                       // lane half (0/1)
    const int nn = lane & 15;                       // N index (row) / M base

    // A operand (16x4 per WMMA): VGPR0 holds K = 2*hi, VGPR1 holds K = 2*hi+1
    v2f Am[4];
#pragma unroll
    for (int q = 0; q < 4; ++q) {
        const int k0 = 4 * q + 2 * hi;
        Am[q].x = (k0     <= nn) ? s * __builtin_amdgcn_exp2f((float)(nn - k0)     * lg2a) : 0.0f;
        Am[q].y = (k0 + 1 <= nn) ? s * __builtin_amdgcn_exp2f((float)(nn - k0 - 1) * lg2a) : 0.0f;
    }
    // carry coefficients per accumulator VGPR v: a^(M+1), M = v + 8*hi
    float coefA[8];
#pragma unroll
    for (int v = 0; v < 8; ++v)
        coefA[v] = __builtin_amdgcn_exp2f((float)(v + 8 * hi + 1) * lg2a);

    const uint64_t xbase = (uint64_t)(uintptr_t)x;

    // Issue one tile: 16 rows x 64 floats; b128 covers 2 rows per instruction
    // (lane half selects row parity, 16 lanes x 16B = one 256B row slice).
    auto issue_tile = [&](int tile, int nb) {
        const int t0 = tile * TILE_T;
#pragma unroll
        for (int j = 0; j < 8; ++j) {
            const int row = 2 * j + hi;
            uint32_t ldsoff = (uint32_t)(uintptr_t)&lds[wv][nb][row * TILE_T + 4 * nn];
            uint32_t goff   = (uint32_t)(((uint32_t)(b0 + row) * C_ + c) * (uint32_t)T_
                                         + (uint32_t)(t0 + 4 * nn)) * 4u;
            asm volatile("global_load_async_to_lds_b128 %0, %1, %2"
                         :: "v"(ldsoff), "v"(goff), "s"(xbase) : "memory");
        }
    };

#pragma unroll
    for (int p = 0; p < NBUF; ++p) issue_tile(p, p);   // 32 async ops in flight

    // seed: carry = x[row][0]  =>  a*carry + s*x0 == x0 == smooth[0]
    asm volatile("s_wait_asynccnt 24" ::: "memory");   // tile 0 landed
    float carry = lds[wv][0][nn * TILE_T];

    for (int tile = 0; tile < NT; ++tile) {
        const int nb = tile & (NBUF - 1);
        // wait until this tile's 8 async loads have landed (in-order retire)
        const int rem = (NT - 1) - tile;               // tiles still issued after this one
        if      (rem >= 3) asm volatile("s_wait_asynccnt 24" ::: "memory");
        else if (rem == 2) asm volatile("s_wait_asynccnt 16" ::: "memory");
        else if (rem == 1) asm volatile("s_wait_asynccnt 8"  ::: "memory");
        else               asm volatile("s_wait_asynccnt 0"  ::: "memory");

        float* bf = &lds[wv][nb][0];

#pragma unroll
        for (int cc = 0; cc < CHUNKS; ++cc) {
            const int tb = cc * 16;

            // B operand (4x16 per WMMA): B[k][n] = x[row n][tb + k]
            v2f Bm[4];
#pragma unroll
            for (int q = 0; q < 4; ++q)
                Bm[q] = *(const v2f*)&bf[nn * TILE_T + tb + 4 * q + 2 * hi];

            v8f acc = {0.f, 0.f, 0.f, 0.f, 0.f, 0.f, 0.f, 0.f};
            acc = __builtin_amdgcn_wmma_f32_16x16x4_f32(false, Am[0], false, Bm[0], (short)0, acc, false, false);
            acc = __builtin_amdgcn_wmma_f32_16x16x4_f32(false, Am[1], false, Bm[1], (short)0, acc, false, false);
            acc = __builtin_amdgcn_wmma_f32_16x16x4_f32(false, Am[2], false, Bm[2], (short)0, acc, false, false);
            acc = __builtin_amdgcn_wmma_f32_16x16x4_f32(false, Am[3], false, Bm[3], (short)0, acc, false, false);

            // smooth = L*X + a^{i+1} * carry
            float sm[8];
#pragma unroll
            for (int v = 0; v < 8; ++v) sm[v] = acc[v] + coefA[v] * carry;

            // next carry = smooth[M=15][n]  (VGPR 7 of lanes 16..31)
            carry = __shfl(sm[7], 16 + nn, 32);

            // x in C/D layout: lane holds row nn, t = tb + 8*hi + v
            const float4 x0 = *(const float4*)&bf[nn * TILE_T + tb + 8 * hi];
            const float4 x1 = *(const float4*)&bf[nn * TILE_T + tb + 8 * hi + 4];
            const float xv[8] = {x0.x, x0.y, x0.z, x0.w, x1.x, x1.y, x1.z, x1.w};

            // pcen = exp2(r*log2(x*exp2(-a*log2(sm+eps)) + d)) - d^r
            // raw v_log_f32/v_exp_f32: inputs >= 1e-6 / >= 0.5, never denormal
            float pc[8];
#pragma unroll
            for (int v = 0; v < 8; ++v) {
                const float psinv = __builtin_amdgcn_exp2f(
                    -al * __builtin_amdgcn_logf(sm[v] + EPSF));        // (sm+eps)^-alpha
                const float arg = __fmaf_rn(xv[v], psinv, de);         // x/ps + delta
                pc[v] = __builtin_amdgcn_exp2f(
                    rr * __builtin_amdgcn_logf(arg)) - der;
            }
            // overwrite x tile in-place with pcen
            *(float4*)&bf[nn * TILE_T + tb + 8 * hi]     = make_float4(pc[0], pc[1], pc[2], pc[3]);
            *(float4*)&bf[nn * TILE_T + tb + 8 * hi + 4] = make_float4(pc[4], pc[5], pc[6], pc[7]);
        }

        // coalesced store-out: b128, 2 rows per instruction (512B/instr)
        const int t0 = tile * TILE_T;
#pragma unroll
        for (int j = 0; j < 8; ++j) {
            const int row = 2 * j + hi;
            const float4 v = *(const float4*)&bf[row * TILE_T + 4 * nn];
            *(float4*)&out[((uint64_t)(b0 + row) * C_ + c) * T_ + t0 + 4 * nn] = v;
        }

        const int nxt = tile + NBUF;
        if (nxt < NT) issue_tile(nxt, nb);
    }
}

extern "C" void kernel_launch(void* const* d_in, const int* in_sizes, int n_in,
                              void* d_out, int out_size, void* d_ws, size_t ws_size,
                              hipStream_t stream) {
    const float* x     = (const float*)d_in[0];
    const float* alpha = (const float*)d_in[1];
    const float* delta = (const float*)d_in[2];
    const float* r     = (const float*)d_in[3];
    const float* s     = (const float*)d_in[4];
    float* out = (float*)d_out;

    // 512 wave-tasks (128 channels x 4 batch-groups), 4 waves per block
    dim3 grid(C_ * (B_ / 16) / WPB);   // 128 blocks
    dim3 block(WPB * 32);              // 128 threads
    pcen_wmma_kernel<<<grid, block, 0, stream>>>(x, alpha, delta, r, s, out);
}